// GNN_85581518340340
// MI455X (gfx1250) — compile-verified
//
#include <hip/hip_runtime.h>
#include <hip/hip_bf16.h>

// ---------------------------------------------------------------------------
// Problem constants (from the reference)
// ---------------------------------------------------------------------------
static const int N0 = 60000, N1 = 30000, N2 = 15000;
static const int E1 = 960000, E2 = 480000;
static const int G = 2000, D = 400, C = 20;

typedef __attribute__((ext_vector_type(2))) float v2f;
typedef __attribute__((ext_vector_type(4))) float f4;
typedef __attribute__((ext_vector_type(8))) float v8f;

// ---------------------------------------------------------------------------
// WMMA f32 GEMM:  Cmat[M,N] = A[M,K] @ B[K,N] + bias[N]
//   A row-major, B K-major (K rows of N floats), fp32 everywhere.
//   Workgroup tile 128x64, K-stage 32, double-buffered LDS (one barrier per
//   stage). 8 waves, each wave owns a 32x32 patch = 2x2 fragments of
//   v_wmma_f32_16x16x4_f32.  B is staged TRANSPOSED in LDS so each B fragment
//   is a single contiguous ds_load_b64.  Requires K % 32 == 0.
// ---------------------------------------------------------------------------
#define BM 128
#define BN 64
#define BK 32
#define LDA  (BK + 4)   // As row stride (floats): 16B-aligned, conflict-free
#define LDBK (BK + 4)   // Bs column stride (floats), Bs[n][k]

__launch_bounds__(256)
__global__ void wmma_gemm_f32(const float* __restrict__ A,
                              const float* __restrict__ B,
                              const float* __restrict__ bias,
                              float* __restrict__ Cmat,
                              int M, int N, int K)
{
    __shared__ float As[2][BM * LDA];    // 2 x 18432 B
    __shared__ float Bs[2][BN * LDBK];   // 2 x  9216 B   (total 55296 B)

    const int tid  = threadIdx.x;
    const int lane = tid & 31;
    const int wave = tid >> 5;            // 0..7
    const int wm   = (wave >> 1) * 32;    // wave M offset in WG tile
    const int wn   = (wave & 1) * 32;     // wave N offset in WG tile
    const int half = lane >> 4;           // 0: K pair {0,1}; 1: K pair {2,3}
    const int l15  = lane & 15;

    const int tileM = blockIdx.x * BM;
    const int tileN = blockIdx.y * BN;

    // global staging coordinates (per thread)
    const int ar  = tid >> 3;             // 0..31   (A row group)
    const int ac  = (tid & 7) * 4;        // 0..28   (A col, float4)
    const int bkr = tid >> 4;             // 0..15   (B k-row group)
    const int bc  = (tid & 15) * 4;       // 0..60   (B col, float4)

    v8f acc[2][2] = {};
    f4  aReg[4], bReg[2];

    // ---- tile loaders / stores ----
    auto load_tile = [&](int k0) {
        #pragma unroll
        for (int i = 0; i < 4; ++i) {
            int gm = tileM + ar + i * 32;
            f4 v = {0.f, 0.f, 0.f, 0.f};
            if (gm < M) v = *(const f4*)(A + (size_t)gm * K + (k0 + ac));
            aReg[i] = v;
        }
        #pragma unroll
        for (int i = 0; i < 2; ++i) {
            int krr = bkr + i * 16;
            int gn  = tileN + bc;
            size_t goff = (size_t)(k0 + krr) * N + gn;
            f4 v = {0.f, 0.f, 0.f, 0.f};
            if (gn + 3 < N && (goff & 3) == 0) {
                v = *(const f4*)(B + goff);
            } else {
                float* pv = (float*)&v;
                #pragma unroll
                for (int j = 0; j < 4; ++j)
                    if (gn + j < N) pv[j] = B[goff + j];
            }
            bReg[i] = v;
        }
    };
    auto store_tile = [&](int buf) {
        #pragma unroll
        for (int i = 0; i < 4; ++i)
            *(f4*)(&As[buf][(ar + i * 32) * LDA + ac]) = aReg[i];
        #pragma unroll
        for (int i = 0; i < 2; ++i) {
            int krr = bkr + i * 16;
            const float* pv = (const float*)&bReg[i];
            #pragma unroll
            for (int j = 0; j < 4; ++j)
                Bs[buf][(bc + j) * LDBK + krr] = pv[j];   // transposed store
        }
    };

    load_tile(0);
    store_tile(0);
    __syncthreads();

    int cur = 0;
    for (int k0 = 0; k0 < K; k0 += BK) {
        const bool hasNext = (k0 + BK) < K;
        if (hasNext) {
            load_tile(k0 + BK);                       // hidden under WMMAs
            if (k0 + 2 * BK < K) {                    // warm GL2 for A stream
                #pragma unroll
                for (int i = 0; i < 4; ++i) {
                    int gm = tileM + ar + i * 32;
                    if (gm < M)
                        __builtin_prefetch(A + (size_t)gm * K + (k0 + 2 * BK + ac), 0, 1);
                }
            }
        }

        const float* as = As[cur];
        const float* bs = Bs[cur];
        #pragma unroll
        for (int kk = 0; kk < BK; kk += 4) {
            const int kc = kk + half * 2;
            v2f afrag[2], bfrag[2];
            #pragma unroll
            for (int i = 0; i < 2; ++i)
                afrag[i] = *(const v2f*)(as + (wm + i * 16 + l15) * LDA + kc);
            #pragma unroll
            for (int j = 0; j < 2; ++j)
                bfrag[j] = *(const v2f*)(bs + (wn + j * 16 + l15) * LDBK + kc);
            #pragma unroll
            for (int i = 0; i < 2; ++i)
                #pragma unroll
                for (int j = 0; j < 2; ++j)
                    acc[i][j] = __builtin_amdgcn_wmma_f32_16x16x4_f32(
                        false, afrag[i], false, bfrag[j],
                        (short)0, acc[i][j], false, false);
        }

        if (hasNext) {
            store_tile(cur ^ 1);
            __syncthreads();
            cur ^= 1;
        }
    }

    // ---- store C (+bias) ----
    #pragma unroll
    for (int j = 0; j < 2; ++j) {
        int col = tileN + wn + j * 16 + l15;
        if (col >= N) continue;
        float bv = bias ? bias[col] : 0.f;
        #pragma unroll
        for (int i = 0; i < 2; ++i) {
            #pragma unroll
            for (int r = 0; r < 8; ++r) {
                int row = tileM + wm + i * 16 + half * 8 + r;
                if (row < M)
                    Cmat[(size_t)row * N + col] = acc[i][j][r] + bv;
            }
        }
    }
}

// ---------------------------------------------------------------------------
// Build K-major concatenated input weight (2000 x 400) + fused bias.
// ---------------------------------------------------------------------------
__global__ void build_bcat(const float* __restrict__ W_sh, const float* __restrict__ W_hu,
                           const float* __restrict__ W_mo, const float* __restrict__ b_sh,
                           const float* __restrict__ b_hu, const float* __restrict__ b_mo,
                           float* __restrict__ Bcat, float* __restrict__ bcat)
{
    int k = blockIdx.x;  // 0..1999
    for (int n = threadIdx.x; n < D; n += blockDim.x) {
        float v;
        if (k < 1000)      v = W_sh[n * 1000 + k];
        else if (k < 1500) v = W_hu[n * 500 + (k - 1000)];
        else               v = W_mo[n * 500 + (k - 1500)];
        Bcat[k * D + n] = v;
    }
    if (blockIdx.x == 0)
        for (int n = threadIdx.x; n < D; n += blockDim.x)
            bcat[n] = b_sh[n] + b_hu[n] + b_mo[n];
}

// W (Nout x K) row-major  ->  B (K x Nout) K-major
__global__ void transpose_w(const float* __restrict__ W, float* __restrict__ B,
                            int Nout, int K)
{
    int idx = blockIdx.x * blockDim.x + threadIdx.x;
    if (idx >= Nout * K) return;
    int k = idx / Nout, n = idx % Nout;
    B[idx] = W[n * K + k];
}

// ---------------------------------------------------------------------------
// Edge scatter: neigh[dst] += h[src] * w_all[eid];  deg[dst] += 1
// One workgroup (128 threads) per edge; 100 lanes each handle 4 features.
// ---------------------------------------------------------------------------
__global__ void scatter_edges(const float* __restrict__ hsrc,
                              const float* __restrict__ w_all,
                              const int* __restrict__ src, const int* __restrict__ dst,
                              const int* __restrict__ eid,
                              float* __restrict__ neigh, float* __restrict__ deg)
{
    int e = blockIdx.x;
    int s = src[e], dd = dst[e];
    float w = w_all[eid[e]];
    if (threadIdx.x < D / 4) {
        const f4 hv = *(const f4*)(hsrc + (size_t)s * D + threadIdx.x * 4);
        float* nb = neigh + (size_t)dd * D + threadIdx.x * 4;
        atomicAdd(nb + 0, hv.x * w);
        atomicAdd(nb + 1, hv.y * w);
        atomicAdd(nb + 2, hv.z * w);
        atomicAdd(nb + 3, hv.w * w);
    }
    if (threadIdx.x == 127) atomicAdd(deg + dd, 1.0f);
}

// neigh[row][:] *= 1 / (deg[row] + 1)
__global__ void finalize_neigh(float* __restrict__ neigh, const float* __restrict__ deg)
{
    int row = blockIdx.x;
    float inv = 1.0f / (deg[row] + 1.0f);
    float* p = neigh + (size_t)row * D;
    for (int f = threadIdx.x; f < D; f += blockDim.x) p[f] *= inv;
}

// ---------------------------------------------------------------------------
// BatchNorm stats over 15000 rows x 50 cols: one block per column.
// ---------------------------------------------------------------------------
__global__ void bn_stats(const float* __restrict__ dmat, float* __restrict__ mean,
                         float* __restrict__ rstd, int R)
{
    __shared__ float s1[256], s2[256];
    int c = blockIdx.x;  // 0..49
    float a = 0.f, b = 0.f;
    for (int r = threadIdx.x; r < R; r += blockDim.x) {
        float v = dmat[(size_t)r * 50 + c];
        a += v; b += v * v;
    }
    s1[threadIdx.x] = a; s2[threadIdx.x] = b;
    __syncthreads();
    for (int s = 128; s > 0; s >>= 1) {
        if ((int)threadIdx.x < s) {
            s1[threadIdx.x] += s1[threadIdx.x + s];
            s2[threadIdx.x] += s2[threadIdx.x + s];
        }
        __syncthreads();
    }
    if (threadIdx.x == 0) {
        float mu  = s1[0] / R;
        float var = s2[0] / R - mu * mu;
        mean[c] = mu;
        rstd[c] = rsqrtf(var + 1e-5f);
    }
}

// Fused BN -> gamma/beta -> ReLU -> (50 -> 2) head
__global__ void bn_relu_head(const float* __restrict__ dmat,
                             const float* __restrict__ mean, const float* __restrict__ rstd,
                             const float* __restrict__ gamma, const float* __restrict__ beta,
                             const float* __restrict__ Wd2, const float* __restrict__ bd2,
                             float* __restrict__ out, int R)
{
    int r = blockIdx.x * blockDim.x + threadIdx.x;
    if (r >= R) return;
    float a0 = bd2[0], a1 = bd2[1];
    const float* row = dmat + (size_t)r * 50;
    #pragma unroll 10
    for (int c = 0; c < 50; ++c) {
        float v = (row[c] - mean[c]) * rstd[c] * gamma[c] + beta[c];
        v = fmaxf(v, 0.f);
        a0 += v * Wd2[c];
        a1 += v * Wd2[50 + c];
    }
    out[(size_t)r * 2 + 0] = a0;
    out[(size_t)r * 2 + 1] = a1;
}

// ---------------------------------------------------------------------------
// Host-side orchestration
// ---------------------------------------------------------------------------
static inline dim3 gemm_grid(int M, int N) {
    return dim3((M + BM - 1) / BM, (N + BN - 1) / BN);
}

extern "C" void kernel_launch(void* const* d_in, const int* in_sizes, int n_in,
                              void* d_out, int out_size, void* d_ws, size_t ws_size,
                              hipStream_t stream)
{
    const float* x     = (const float*)d_in[0];
    const float* w_all = (const float*)d_in[1];
    const float* W_sh  = (const float*)d_in[2];
    const float* b_sh  = (const float*)d_in[3];
    const float* W_hu  = (const float*)d_in[4];
    const float* b_hu  = (const float*)d_in[5];
    const float* W_mo  = (const float*)d_in[6];
    const float* b_mo  = (const float*)d_in[7];
    const float* Wn1   = (const float*)d_in[8];
    const float* bn1   = (const float*)d_in[9];
    const float* Wn2   = (const float*)d_in[10];
    const float* bn2   = (const float*)d_in[11];
    const float* W_lin = (const float*)d_in[12];
    const float* b_lin = (const float*)d_in[13];
    const float* Wd1   = (const float*)d_in[14];
    const float* bd1   = (const float*)d_in[15];
    const float* gamma = (const float*)d_in[16];
    const float* beta  = (const float*)d_in[17];
    const float* Wd2   = (const float*)d_in[18];
    const float* bd2   = (const float*)d_in[19];
    const int* src1 = (const int*)d_in[20];
    const int* dst1 = (const int*)d_in[21];
    const int* eid1 = (const int*)d_in[22];
    const int* src2 = (const int*)d_in[23];
    const int* dst2 = (const int*)d_in[24];
    const int* eid2 = (const int*)d_in[25];

    float* out = (float*)d_out;       // [0,300000): h_x ; [300000,330000): domain_x

    // -------- workspace layout (floats) --------
    float* ws     = (float*)d_ws;
    float* Bcat   = ws;                          // 2000*400
    float* bcat   = Bcat   + (size_t)G * D;      // 400
    float* Bn1    = bcat   + D;                  // 400*400
    float* Bn2    = Bn1    + (size_t)D * D;      // 400*400
    float* Blin   = Bn2    + (size_t)D * D;      // 400*20
    float* Bd1    = Blin   + (size_t)D * C;      // 400*50
    float* h      = Bd1    + (size_t)D * 50;     // 60000*400
    float* neigh1 = h      + (size_t)N0 * D;     // 30000*400
    float* deg1   = neigh1 + (size_t)N1 * D;     // 30000
    float* h1     = deg1   + N1;                 // 30000*400
    float* neigh2 = h1     + (size_t)N1 * D;     // 15000*400
    float* deg2   = neigh2 + (size_t)N2 * D;     // 15000
    float* h2     = deg2   + N2;                 // 15000*400
    float* dbuf   = h2     + (size_t)N2 * D;     // 15000*50
    float* mean   = dbuf   + (size_t)N2 * 50;    // 50
    float* rstd   = mean   + 50;                 // 50

    // -------- weight prep --------
    build_bcat<<<G, 256, 0, stream>>>(W_sh, W_hu, W_mo, b_sh, b_hu, b_mo, Bcat, bcat);
    transpose_w<<<(D * D + 255) / 256, 256, 0, stream>>>(Wn1, Bn1, D, D);
    transpose_w<<<(D * D + 255) / 256, 256, 0, stream>>>(Wn2, Bn2, D, D);
    transpose_w<<<(C * D + 255) / 256, 256, 0, stream>>>(W_lin, Blin, C, D);
    transpose_w<<<(50 * D + 255) / 256, 256, 0, stream>>>(Wd1, Bd1, 50, D);

    // -------- h = x @ Bcat + bcat   (M=60000, N=400, K=2000) --------
    wmma_gemm_f32<<<gemm_grid(N0, D), 256, 0, stream>>>(x, Bcat, bcat, h, N0, D, G);

    // -------- GNN layer 1 --------
    hipMemcpyAsync(neigh1, h, (size_t)N1 * D * sizeof(float),
                   hipMemcpyDeviceToDevice, stream);
    hipMemsetAsync(deg1, 0, (size_t)N1 * sizeof(float), stream);
    scatter_edges<<<E1, 128, 0, stream>>>(h, w_all, src1, dst1, eid1, neigh1, deg1);
    finalize_neigh<<<N1, 128, 0, stream>>>(neigh1, deg1);
    wmma_gemm_f32<<<gemm_grid(N1, D), 256, 0, stream>>>(neigh1, Bn1, bn1, h1, N1, D, D);

    // -------- GNN layer 2 --------
    hipMemcpyAsync(neigh2, h1, (size_t)N2 * D * sizeof(float),
                   hipMemcpyDeviceToDevice, stream);
    hipMemsetAsync(deg2, 0, (size_t)N2 * sizeof(float), stream);
    scatter_edges<<<E2, 128, 0, stream>>>(h1, w_all, src2, dst2, eid2, neigh2, deg2);
    finalize_neigh<<<N2, 128, 0, stream>>>(neigh2, deg2);
    wmma_gemm_f32<<<gemm_grid(N2, D), 256, 0, stream>>>(neigh2, Bn2, bn2, h2, N2, D, D);

    // -------- heads --------
    // h_x = h2 @ W_lin^T + b_lin  -> out[0 : 15000*20]
    wmma_gemm_f32<<<gemm_grid(N2, C), 256, 0, stream>>>(h2, Blin, b_lin, out, N2, C, D);
    // d = h2 @ Wd1^T + bd1
    wmma_gemm_f32<<<gemm_grid(N2, 50), 256, 0, stream>>>(h2, Bd1, bd1, dbuf, N2, 50, D);
    // BN stats + fused BN/ReLU/(50->2)
    bn_stats<<<50, 256, 0, stream>>>(dbuf, mean, rstd, N2);
    bn_relu_head<<<(N2 + 255) / 256, 256, 0, stream>>>(
        dbuf, mean, rstd, gamma, beta, Wd2, bd2, out + (size_t)N2 * C, N2);
}